// RelationalDistanceLoss_69647189671965
// MI455X (gfx1250) — compile-verified
//
#include <hip/hip_runtime.h>
#include <hip/hip_bf16.h>

typedef __attribute__((ext_vector_type(2))) float v2f;
typedef __attribute__((ext_vector_type(8))) float v8f;

#define B_N 256
#define D_K 768
#define MARGIN_F 0.2f

// ---------------------------------------------------------------------------
// Kernel 1: Gram matrix G = X * X^T via V_WMMA_F32_16X16X4_F32.
// One wave (32 lanes) computes one 16x16 output tile, iterating K in steps of 4.
// A-operand layout (16x4 f32, 2 VGPRs): lane L, vgpr v -> A[L&15][2*(L>>4)+v]
// B-operand layout (4x16 f32, 2 VGPRs): lane L, vgpr v -> B[2*(L>>4)+v][L&15]
// Since B = X^T, both operands use the identical per-lane load pattern from X,
// differing only in the row-block base. 8 waves per 256-thread block.
// ---------------------------------------------------------------------------
__global__ void __launch_bounds__(256)
gram_wmma_f32(const float* __restrict__ X, float* __restrict__ G) {
    const int lane = threadIdx.x & 31;
    const int wave = threadIdx.x >> 5;
    const int tile = blockIdx.x * 8 + wave;      // 0..255
    const int tr = (tile >> 4) << 4;             // tile row base
    const int tc = (tile & 15) << 4;             // tile col base

    const int m    = lane & 15;                  // M (and N) index within tile
    const int koff = (lane >> 4) << 1;           // 0 or 2

    const float* __restrict__ rowA = X + (size_t)(tr + m) * D_K + koff;
    const float* __restrict__ rowB = X + (size_t)(tc + m) * D_K + koff;

    v8f c = {};
#pragma unroll 4
    for (int k = 0; k < D_K; k += 4) {
        v2f a = *(const v2f*)(rowA + k);
        v2f b = *(const v2f*)(rowB + k);
        // D = A * B + C ; f32 in, f32 accumulate (full precision Gram)
        c = __builtin_amdgcn_wmma_f32_16x16x4_f32(
                /*neg_a=*/false, a, /*neg_b=*/false, b,
                /*c_mod=*/(short)0, c, /*reuse_a=*/false, /*reuse_b=*/false);
    }

    // C/D layout: vgpr v, lane L -> (M = v + 8*(L>>4), N = L&15)
    const int mhalf = (lane >> 4) << 3;
#pragma unroll
    for (int v = 0; v < 8; ++v) {
        G[(size_t)(tr + mhalf + v) * B_N + (tc + m)] = c[v];
    }
}

// ---------------------------------------------------------------------------
// Kernel 2: distances from Gram: d[i][j] = sqrt(max(G[i][i]+G[j][j]-2G[i][j],0))
// ---------------------------------------------------------------------------
__global__ void __launch_bounds__(256)
dist_from_gram(const float* __restrict__ G, float* __restrict__ Dm) {
    const int idx = blockIdx.x * blockDim.x + threadIdx.x;  // 65536 threads
    const int i = idx >> 8;
    const int j = idx & 255;
    const float sq_i = G[(size_t)i * B_N + i];
    const float sq_j = G[(size_t)j * B_N + j];
    float d2 = sq_i + sq_j - 2.0f * G[(size_t)i * B_N + j];
    d2 = fmaxf(d2, 0.0f);
    Dm[idx] = (d2 > 0.0f) ? sqrtf(d2) : 0.0f;
}

// ---------------------------------------------------------------------------
// Kernel 3: triplet loss. Block = one anchor i. Stage distance rows in LDS;
// thread t owns j = t, loops over k > j. Deterministic block tree reduction.
// ---------------------------------------------------------------------------
__global__ void __launch_bounds__(256)
triplet_loss_rows(const float* __restrict__ De, const float* __restrict__ Dg,
                  float* __restrict__ partial) {
    __shared__ float se[B_N];
    __shared__ float sg[B_N];
    __shared__ float red[B_N];

    const int i = blockIdx.x;
    const int t = threadIdx.x;

    se[t] = De[(size_t)i * B_N + t];
    sg[t] = Dg[(size_t)i * B_N + t];
    __syncthreads();

    float acc = 0.0f;
    const int j = t;
    if (j != i) {
        const float e_ij = se[j];
        const float g_ij = sg[j];
        for (int k = j + 1; k < B_N; ++k) {
            if (k == i) continue;
            const float e_ik = se[k];
            const float g_ik = sg[k];
            const float diff = g_ij - g_ik;
            float l;
            if (e_ij < e_ik) {
                l = fmaxf(diff + MARGIN_F, 0.0f);
            } else if (e_ik < e_ij) {
                l = fmaxf(MARGIN_F - diff, 0.0f);
            } else {
                l = fabsf(diff);
            }
            acc += l;
        }
    }
    red[t] = acc;
    __syncthreads();
#pragma unroll
    for (int s = 128; s > 0; s >>= 1) {
        if (t < s) red[t] += red[t + s];
        __syncthreads();
    }
    if (t == 0) partial[i] = red[0];
}

// ---------------------------------------------------------------------------
// Kernel 4: deterministic final reduction in double, scale, write scalar.
// ---------------------------------------------------------------------------
__global__ void finalize_loss(const float* __restrict__ partial,
                              float* __restrict__ out) {
    double s = 0.0;
    for (int i = 0; i < B_N; ++i) s += (double)partial[i];
    const double num_triplets = (double)B_N * (B_N - 1) * (B_N - 2) * 0.5;
    out[0] = (float)(s / num_triplets);
}

extern "C" void kernel_launch(void* const* d_in, const int* in_sizes, int n_in,
                              void* d_out, int out_size, void* d_ws, size_t ws_size,
                              hipStream_t stream) {
    (void)in_sizes; (void)n_in; (void)out_size; (void)ws_size;
    const float* img = (const float*)d_in[0];  // (256, 768) f32
    const float* ehr = (const float*)d_in[1];  // (256, 768) f32

    float* ws      = (float*)d_ws;
    float* G_img   = ws;                 // 65536 f32
    float* G_ehr   = ws + 65536;         // 65536 f32
    float* D_img   = ws + 131072;        // 65536 f32
    float* D_ehr   = ws + 196608;        // 65536 f32
    float* partial = ws + 262144;        // 256 f32

    gram_wmma_f32<<<32, 256, 0, stream>>>(img, G_img);
    gram_wmma_f32<<<32, 256, 0, stream>>>(ehr, G_ehr);

    dist_from_gram<<<256, 256, 0, stream>>>(G_img, D_img);
    dist_from_gram<<<256, 256, 0, stream>>>(G_ehr, D_ehr);

    triplet_loss_rows<<<256, 256, 0, stream>>>(D_ehr, D_img, partial);

    finalize_loss<<<1, 1, 0, stream>>>(partial, (float*)d_out);
}